// FFT2_77472620085500
// MI455X (gfx1250) — compile-verified
//
#include <hip/hip_runtime.h>
#include <hip/hip_bf16.h>

// FFT-mixer block for MI455X (gfx1250, wave32).
// Stage 1/4 use V_WMMA_F32_16X16X4_F32 (fp32 matrix pipe, matches fp32 ref).
// Stage 2 = per-patch 8x8 rFFT2 / spectral filter / GELU / irFFT2 (VALU).
// Stage 3 = 7x7 depthwise conv via LDS halo tile + exact GELU.

typedef float v2f __attribute__((ext_vector_type(2)));
typedef float v8f __attribute__((ext_vector_type(8)));

#define HW 65536   // 256*256
#define NB 8
#define C1 32
#define C2 64

__device__ __forceinline__ v8f wmma_f32x4(v2f a, v2f b, v8f c) {
  // 8 args: (neg_a, A, neg_b, B, c_mod, C, reuse_a, reuse_b)
  return __builtin_amdgcn_wmma_f32_16x16x4_f32(false, a, false, b, (short)0, c,
                                               false, false);
}

__device__ __forceinline__ float gelu_exact(float v) {
  return 0.5f * v * (1.0f + erff(v * 0.70710678118654752f));
}

// ---------------------------------------------------------------- stage 1
// out[b, co, p] = sum_ci w[co,ci] * x[b,ci,p] + bias[co]   (M=64,K=32,N=BHW)
// wave: 64x16 output tile (4 M-tiles of 16), K in steps of 4 via f32 WMMA.
__global__ void k_conv_in(const float* __restrict__ x, const float* __restrict__ w,
                          const float* __restrict__ bias, float* __restrict__ out) {
  const int wave = threadIdx.x >> 5;
  const int lane = threadIdx.x & 31;
  const int n0   = blockIdx.x * 128 + wave * 16;   // pixel over B*HW
  const int b    = n0 >> 16;
  const int pix  = n0 & (HW - 1);
  const int col  = lane & 15;
  const int klo  = (lane >> 4) << 1;               // 0 (lanes 0-15) / 2 (16-31)
  const float* xb = x + (size_t)b * C1 * HW + pix + col;

  v8f acc0 = {}, acc1 = {}, acc2 = {}, acc3 = {};
#pragma unroll
  for (int kk = 0; kk < C1; kk += 4) {
    v2f bm;                                        // B: 4x16 slice of x
    bm.x = xb[(size_t)(kk + klo) * HW];
    bm.y = xb[(size_t)(kk + klo + 1) * HW];
    v2f a0, a1, a2, a3;                            // A: 16x4 slices of w
    a0.x = w[(col +  0) * C1 + kk + klo]; a0.y = w[(col +  0) * C1 + kk + klo + 1];
    a1.x = w[(col + 16) * C1 + kk + klo]; a1.y = w[(col + 16) * C1 + kk + klo + 1];
    a2.x = w[(col + 32) * C1 + kk + klo]; a2.y = w[(col + 32) * C1 + kk + klo + 1];
    a3.x = w[(col + 48) * C1 + kk + klo]; a3.y = w[(col + 48) * C1 + kk + klo + 1];
    acc0 = wmma_f32x4(a0, bm, acc0);
    acc1 = wmma_f32x4(a1, bm, acc1);
    acc2 = wmma_f32x4(a2, bm, acc2);
    acc3 = wmma_f32x4(a3, bm, acc3);
  }
  const int mo = (lane >> 4) << 3;                 // C/D: M = v + 8*(lane>=16)
  float* ob = out + (size_t)b * C2 * HW + pix + col;
#pragma unroll
  for (int v = 0; v < 8; ++v) {
    int r0 = v + mo, r1 = 16 + v + mo, r2 = 32 + v + mo, r3 = 48 + v + mo;
    ob[(size_t)r0 * HW] = acc0[v] + bias[r0];
    ob[(size_t)r1 * HW] = acc1[v] + bias[r1];
    ob[(size_t)r2 * HW] = acc2[v] + bias[r2];
    ob[(size_t)r3 * HW] = acc3[v] + bias[r3];
  }
}

// ---------------------------------------------------------------- stage 2
// One thread per 8x8 patch: rfft2 -> *fft1 -> gelu(re),gelu(im) -> *fft2 -> irfft2.
__global__ void k_fft_block(const float* __restrict__ in, const float* __restrict__ f1,
                            const float* __restrict__ f2, float* __restrict__ out) {
  const float r = 0.70710678118654752f;
  const float c8[8] = {1.f, r, 0.f, -r, -1.f, -r, 0.f, r};   // cos(2*pi*k/8)
  const float s8[8] = {0.f, r, 1.f, r, 0.f, -r, -1.f, -r};   // sin(2*pi*k/8)

  const int idx   = blockIdx.x * blockDim.x + threadIdx.x;   // 0..524287
  const int clin  = idx >> 10;                               // b*64 + c
  const int pp    = idx & 1023;
  const int ph    = pp >> 5, pw = pp & 31;
  const int c     = clin & (C2 - 1);
  const size_t base = (size_t)clin * HW + (size_t)(ph * 8) * 256 + pw * 8;
  const float* src = in + base;
  float* dst = out + base;

  float Rre[8][5], Rim[8][5];
  // row-wise rfft8 (real -> 5 complex bins)
#pragma unroll
  for (int h = 0; h < 8; ++h) {
    float xr[8];
#pragma unroll
    for (int w = 0; w < 8; ++w) xr[w] = src[h * 256 + w];
#pragma unroll
    for (int v = 0; v < 5; ++v) {
      float re = 0.f, im = 0.f;
#pragma unroll
      for (int w = 0; w < 8; ++w) {
        int t = (w * v) & 7;
        re += xr[w] * c8[t];
        im -= xr[w] * s8[t];
      }
      Rre[h][v] = re; Rim[h][v] = im;
    }
  }
  const float* g1 = f1 + c * 40;   // [8][5] per channel
  const float* g2 = f2 + c * 40;
  // per spectral column: full fft8 along h, filter/gelu/filter, ifft8 back
#pragma unroll
  for (int v = 0; v < 5; ++v) {
    float Fre[8], Fim[8];
#pragma unroll
    for (int u = 0; u < 8; ++u) {
      float re = 0.f, im = 0.f;
#pragma unroll
      for (int h = 0; h < 8; ++h) {
        int t = (u * h) & 7;
        re += Rre[h][v] * c8[t] + Rim[h][v] * s8[t];
        im += Rim[h][v] * c8[t] - Rre[h][v] * s8[t];
      }
      float g = g1[u * 5 + v];
      re = gelu_exact(re * g);
      im = gelu_exact(im * g);
      g = g2[u * 5 + v];
      Fre[u] = re * g; Fim[u] = im * g;
    }
#pragma unroll
    for (int h = 0; h < 8; ++h) {     // inverse fft along h (unnormalized)
      float re = 0.f, im = 0.f;
#pragma unroll
      for (int u = 0; u < 8; ++u) {
        int t = (u * h) & 7;
        re += Fre[u] * c8[t] - Fim[u] * s8[t];
        im += Fim[u] * c8[t] + Fre[u] * s8[t];
      }
      Rre[h][v] = re; Rim[h][v] = im;
    }
  }
  // row-wise irfft8 (imag of DC/Nyquist ignored, per numpy semantics); 1/64 norm
#pragma unroll
  for (int h = 0; h < 8; ++h) {
#pragma unroll
    for (int w = 0; w < 8; ++w) {
      float acc = Rre[h][0] + ((w & 1) ? -Rre[h][4] : Rre[h][4]);
#pragma unroll
      for (int v = 1; v < 4; ++v) {
        int t = (v * w) & 7;
        acc += 2.f * (Rre[h][v] * c8[t] - Rim[h][v] * s8[t]);
      }
      dst[h * 256 + w] = acc * 0.015625f;
    }
  }
}

// ---------------------------------------------------------------- stage 3
// 7x7 depthwise conv (pad 3) + exact GELU. 32x8 output tile, 38x14 LDS halo.
__global__ void k_dwconv(const float* __restrict__ in, const float* __restrict__ w,
                         const float* __restrict__ bias, float* __restrict__ out) {
  __shared__ float tile[14 * 38];
  __shared__ float wk[49];
  const int p  = blockIdx.y;                 // b*64 + c
  const int c  = p & (C2 - 1);
  const int tX = (blockIdx.x & 7) * 32;
  const int tY = (blockIdx.x >> 3) * 8;
  const int tid = threadIdx.x;
  if (tid < 49) wk[tid] = w[c * 49 + tid];
  const float* plane = in + (size_t)p * HW;
  for (int i = tid; i < 14 * 38; i += 256) {
    int rr = i / 38, cc = i - rr * 38;
    int gy = tY + rr - 3, gx = tX + cc - 3;
    float v = 0.f;
    if (gy >= 0 && gy < 256 && gx >= 0 && gx < 256) v = plane[gy * 256 + gx];
    tile[i] = v;
  }
  __syncthreads();
  const int tx = tid & 31, ty = tid >> 5;
  float acc = bias[c];
#pragma unroll
  for (int ky = 0; ky < 7; ++ky)
#pragma unroll
    for (int kx = 0; kx < 7; ++kx)
      acc += wk[ky * 7 + kx] * tile[(ty + ky) * 38 + (tx + kx)];
  out[(size_t)p * HW + (tY + ty) * 256 + (tX + tx)] = gelu_exact(acc);
}

// ---------------------------------------------------------------- stage 4
// out[b, co, p] = sum_ci w[co,ci] * x[b,ci,p] + bias[co]   (M=32,K=64,N=BHW)
__global__ void k_conv_out(const float* __restrict__ x, const float* __restrict__ w,
                           const float* __restrict__ bias, float* __restrict__ out) {
  const int wave = threadIdx.x >> 5;
  const int lane = threadIdx.x & 31;
  const int n0   = blockIdx.x * 128 + wave * 16;
  const int b    = n0 >> 16;
  const int pix  = n0 & (HW - 1);
  const int col  = lane & 15;
  const int klo  = (lane >> 4) << 1;
  const float* xb = x + (size_t)b * C2 * HW + pix + col;

  v8f acc0 = {}, acc1 = {};
#pragma unroll
  for (int kk = 0; kk < C2; kk += 4) {
    v2f bm;
    bm.x = xb[(size_t)(kk + klo) * HW];
    bm.y = xb[(size_t)(kk + klo + 1) * HW];
    v2f a0, a1;
    a0.x = w[(col +  0) * C2 + kk + klo]; a0.y = w[(col +  0) * C2 + kk + klo + 1];
    a1.x = w[(col + 16) * C2 + kk + klo]; a1.y = w[(col + 16) * C2 + kk + klo + 1];
    acc0 = wmma_f32x4(a0, bm, acc0);
    acc1 = wmma_f32x4(a1, bm, acc1);
  }
  const int mo = (lane >> 4) << 3;
  float* ob = out + (size_t)b * C1 * HW + pix + col;
#pragma unroll
  for (int v = 0; v < 8; ++v) {
    int r0 = v + mo, r1 = 16 + v + mo;
    ob[(size_t)r0 * HW] = acc0[v] + bias[r0];
    ob[(size_t)r1 * HW] = acc1[v] + bias[r1];
  }
}

extern "C" void kernel_launch(void* const* d_in, const int* in_sizes, int n_in,
                              void* d_out, int out_size, void* d_ws, size_t ws_size,
                              hipStream_t stream) {
  (void)in_sizes; (void)n_in; (void)out_size; (void)ws_size;
  const float* x     = (const float*)d_in[0];
  const float* w_in  = (const float*)d_in[1];
  const float* b_in  = (const float*)d_in[2];
  const float* fft1  = (const float*)d_in[3];
  const float* fft2  = (const float*)d_in[4];
  const float* w_dw  = (const float*)d_in[5];
  const float* b_dw  = (const float*)d_in[6];
  const float* w_out = (const float*)d_in[7];
  const float* b_out = (const float*)d_in[8];

  float* bufA = (float*)d_ws;                          // [8,64,256,256]
  float* bufB = bufA + (size_t)NB * C2 * HW;           // [8,64,256,256]

  k_conv_in  <<<4096, 256, 0, stream>>>(x, w_in, b_in, bufA);
  k_fft_block<<<2048, 256, 0, stream>>>(bufA, fft1, fft2, bufB);
  k_dwconv   <<<dim3(256, 512), 256, 0, stream>>>(bufB, w_dw, b_dw, bufA);
  k_conv_out <<<4096, 256, 0, stream>>>(bufA, w_out, b_out, (float*)d_out);
}